// GCNConvLayer_6820408066750
// MI455X (gfx1250) — compile-verified
//
#include <hip/hip_runtime.h>

#define N_NODES 50000
#define N_EDGES 800000
#define DIM     128

typedef __attribute__((ext_vector_type(2))) float v2f;
typedef __attribute__((ext_vector_type(8))) float v8f;

// ---------------------------------------------------------------------------
// 1) deg[n] = 1.0 (self loop)
// ---------------------------------------------------------------------------
__global__ void init_deg_kernel(float* __restrict__ deg) {
    int i = blockIdx.x * blockDim.x + threadIdx.x;
    if (i < N_NODES) deg[i] = 1.0f;
}

// ---------------------------------------------------------------------------
// 2) deg[dst[e]] += 1
// ---------------------------------------------------------------------------
__global__ void count_deg_kernel(const long long* __restrict__ ei,
                                 float* __restrict__ deg) {
    int e = blockIdx.x * blockDim.x + threadIdx.x;
    if (e < N_EDGES) {
        long long d = ei[(size_t)N_EDGES + e];   // dst row of edge_index
        atomicAdd(&deg[d], 1.0f);
    }
}

// ---------------------------------------------------------------------------
// 3) dinv[n] = rsqrt(deg[n])   (in place)
// ---------------------------------------------------------------------------
__global__ void rsqrt_kernel(float* __restrict__ deg) {
    int i = blockIdx.x * blockDim.x + threadIdx.x;
    if (i < N_NODES) deg[i] = rsqrtf(deg[i]);
}

// ---------------------------------------------------------------------------
// 4) xw = x @ W  via V_WMMA_F32_16X16X4_F32.
//    Each wave: one 16-row stripe x all 128 cols (8 v8f accumulators),
//    K loop of 32 x (8 wmma) = 256 wmma per wave.
// ---------------------------------------------------------------------------
__global__ __launch_bounds__(128) void gemm_wmma_kernel(
        const float* __restrict__ x,
        const float* __restrict__ W,
        float* __restrict__ xw) {
    const int lane   = threadIdx.x & 31;
    const int wave   = threadIdx.x >> 5;
    const int stripe = blockIdx.x * 4 + wave;        // 16 rows per stripe
    if (stripe >= N_NODES / 16) return;              // wave-uniform guard

    const int m0   = stripe * 16;
    const int mRow = lane & 15;                      // A: M index
    const int nIdx = lane & 15;                      // B/D: N index
    const int koff = (lane >> 4) << 1;               // 0 (lanes 0-15) or 2

    v8f acc[8] = {};                                 // 16x128 f32 accum

    const float* __restrict__ arow = x + (size_t)(m0 + mRow) * DIM;

    for (int kb = 0; kb < DIM; kb += 4) {
        // A tile 16x4: VGPR v, lane l -> x[m0+(l&15)][kb + (l>>4)*2 + v]
        v2f a;
        a.x = arow[kb + koff + 0];
        a.y = arow[kb + koff + 1];
#pragma unroll
        for (int t = 0; t < 8; ++t) {
            // B tile 4x16: VGPR v, lane l -> W[kb + (l>>4)*2 + v][t*16 + (l&15)]
            const int col = t * 16 + nIdx;
            v2f bm;
            bm.x = W[(size_t)(kb + koff + 0) * DIM + col];
            bm.y = W[(size_t)(kb + koff + 1) * DIM + col];
            acc[t] = __builtin_amdgcn_wmma_f32_16x16x4_f32(
                false, a, false, bm, (short)0, acc[t], false, false);
        }
    }

    // D layout: VGPR r, lane l -> row m0 + r + (l>>4)*8, col l&15
#pragma unroll
    for (int t = 0; t < 8; ++t) {
        const int col = t * 16 + nIdx;
#pragma unroll
        for (int r = 0; r < 8; ++r) {
            const int row = m0 + r + ((lane >> 4) << 3);
            xw[(size_t)row * DIM + col] = acc[t][r];
        }
    }
}

// ---------------------------------------------------------------------------
// 5) agg[n] = dinv[n]^2 * xw[n]   (self-loop term; also initializes agg)
// ---------------------------------------------------------------------------
__global__ void self_agg_kernel(const float* __restrict__ dinv,
                                const float* __restrict__ xw,
                                float* __restrict__ agg) {
    size_t i = (size_t)blockIdx.x * blockDim.x + threadIdx.x;
    if (i < (size_t)N_NODES * DIM) {
        int n = (int)(i >> 7);                       // /128
        float dv = dinv[n];
        agg[i] = dv * dv * xw[i];
    }
}

// ---------------------------------------------------------------------------
// 6) agg[dst] += dinv[src]*dinv[dst] * xw[src]   (one wave per edge,
//    float4 gather: 32 lanes x 4 floats = 128 dims; xw/agg are L2-resident)
// ---------------------------------------------------------------------------
__global__ __launch_bounds__(256) void edge_scatter_kernel(
        const long long* __restrict__ ei,
        const float* __restrict__ dinv,
        const float* __restrict__ xw,
        float* __restrict__ agg) {
    int e    = (int)(((size_t)blockIdx.x * blockDim.x + threadIdx.x) >> 5);
    int lane = threadIdx.x & 31;
    if (e >= N_EDGES) return;

    long long s = ei[e];
    long long d = ei[(size_t)N_EDGES + e];
    float nrm = dinv[s] * dinv[d];

    const float4* __restrict__ xs = (const float4*)(xw + (size_t)s * DIM);
    float* __restrict__ ad = agg + (size_t)d * DIM + (size_t)lane * 4;

    float4 v = xs[lane];
    atomicAdd(ad + 0, nrm * v.x);
    atomicAdd(ad + 1, nrm * v.y);
    atomicAdd(ad + 2, nrm * v.z);
    atomicAdd(ad + 3, nrm * v.w);
}

// ---------------------------------------------------------------------------
// 7) out = x + relu(agg + b)      (crossover is identity since x2 == x1)
// ---------------------------------------------------------------------------
__global__ void epilogue_kernel(const float* __restrict__ x,
                                const float* __restrict__ b,
                                const float* __restrict__ agg,
                                float* __restrict__ out) {
    size_t i = (size_t)blockIdx.x * blockDim.x + threadIdx.x;
    if (i < (size_t)N_NODES * DIM) {
        float v = agg[i] + b[i & (DIM - 1)];
        out[i] = x[i] + fmaxf(v, 0.0f);
    }
}

// ---------------------------------------------------------------------------
extern "C" void kernel_launch(void* const* d_in, const int* in_sizes, int n_in,
                              void* d_out, int out_size, void* d_ws, size_t ws_size,
                              hipStream_t stream) {
    const float*     x  = (const float*)d_in[0];
    const long long* ei = (const long long*)d_in[1];   // int64 per reference
    const float*     W  = (const float*)d_in[2];
    const float*     b  = (const float*)d_in[3];
    // d_in[4] (mask) intentionally unused: mask*x1 + (1-mask)*x1 == x1 exactly.

    // Workspace layout: dinv (200KB) | xw (25.6MB) | agg (25.6MB) ~ 51.4MB
    char* ws = (char*)d_ws;
    float* deg = (float*)ws;                                        // -> dinv
    size_t off1 = ((size_t)N_NODES * sizeof(float) + 255) & ~(size_t)255;
    float* xw  = (float*)(ws + off1);
    size_t off2 = off1 + (size_t)N_NODES * DIM * sizeof(float);
    float* agg = (float*)(ws + off2);

    const size_t ND = (size_t)N_NODES * DIM;

    init_deg_kernel<<<(N_NODES + 255) / 256, 256, 0, stream>>>(deg);
    count_deg_kernel<<<(N_EDGES + 255) / 256, 256, 0, stream>>>(ei, deg);
    rsqrt_kernel<<<(N_NODES + 255) / 256, 256, 0, stream>>>(deg);

    // 3125 stripes of 16 rows, 4 waves (128 threads) per block
    gemm_wmma_kernel<<<(N_NODES / 16 + 3) / 4, 128, 0, stream>>>(x, W, xw);

    self_agg_kernel<<<(int)((ND + 255) / 256), 256, 0, stream>>>(deg, xw, agg);

    // one wave per edge, 8 waves per block
    edge_scatter_kernel<<<(N_EDGES + 7) / 8, 256, 0, stream>>>(ei, deg, xw, agg);

    epilogue_kernel<<<(int)((ND + 255) / 256), 256, 0, stream>>>(x, b, agg,
                                                                 (float*)d_out);
}